// LocalCluster_6158983102550
// MI455X (gfx1250) — compile-verified
//
#include <hip/hip_runtime.h>
#include <math.h>

typedef __attribute__((ext_vector_type(2))) float v2f;
typedef __attribute__((ext_vector_type(8))) float v8f;

#define HW   112
#define CHS  (HW*HW)     // 12544 channel stride
#define CMID 16
#define CIN  128
#define K49  49

__device__ __forceinline__ v8f wmma4(v2f a, v2f b, v8f c) {
  // V_WMMA_F32_16X16X4_F32 : D = A(16x4) x B(4x16) + C, fp32 throughout
  return __builtin_amdgcn_wmma_f32_16x16x4_f32(
      /*neg_a=*/false, a, /*neg_b=*/false, b,
      /*c_mod=*/(short)0, c, /*reuse_a=*/false, /*reuse_b=*/false);
}

__global__ __launch_bounds__(256)
void local_cluster_fused(const float* __restrict__ x,
                         const float* __restrict__ f_w, const float* __restrict__ f_b,
                         const float* __restrict__ v_w, const float* __restrict__ v_b,
                         const float* __restrict__ p_w, const float* __restrict__ p_b,
                         const float* __restrict__ alpha_p, const float* __restrict__ beta_p,
                         float* __restrict__ outg)
{
  __shared__ float fv_lds[2 * CMID * 256];   // f tile then v tile (16 x 256 each)
  __shared__ float cent[K49 * CMID];         // centers -> normalized in place
  __shared__ float aggb[K49 * CMID];         // value_centers -> agg accumulator
  __shared__ float denom[K49];

  float* f_lds = fv_lds;
  float* v_lds = fv_lds + CMID * 256;

  const int tid  = threadIdx.x;
  const int wave = tid >> 5;
  const int lane = tid & 31;
  const int hl   = lane >> 4;   // half-wave select (K split for wmma frags)
  const int ln   = lane & 15;

  const int t     = blockIdx.x;       // tile id: 16 * 49
  const int batch = t / 49;
  const int rr    = t % 49;
  const int wg    = rr / 7;
  const int hg    = rr % 7;
  const long base = (long)(wg * 16) * HW + hg * 16;

  // ================= Phase A: f = f_w@x1 + f_b ; v = v_w@x2 + v_b ==========
  // waves 0-3 -> f (channels 0..127), waves 4-7 -> v (channels 128..255)
  const bool  isV = wave >= 4;
  const float* Wt = isV ? v_w : f_w;
  const float* bt = isV ? v_b : f_b;
  const float* xb = x + (long)batch * 256 * CHS + (isV ? (long)CIN * CHS : 0) + base;
  float* dst      = isV ? v_lds : f_lds;
  const int nt0   = (wave & 3) * 4;   // 4 spatial-row tiles per wave

  v8f acc0 = {}, acc1 = {}, acc2 = {}, acc3 = {};
  #pragma unroll 2
  for (int k0 = 0; k0 < CIN; k0 += 4) {
    // A frag: lane row m=ln, K = k0+2*hl, k0+2*hl+1 (contiguous -> float2)
    v2f a = *(const v2f*)(Wt + ln * CIN + k0 + 2 * hl);
    const float* pk = xb + (long)(k0 + 2 * hl) * CHS + ln;
    // x is streamed exactly once kernel-wide -> non-temporal, keep weights in L2
    { const float* p = pk + (nt0 + 0) * HW; v2f b;
      b.x = __builtin_nontemporal_load(p); b.y = __builtin_nontemporal_load(p + CHS);
      acc0 = wmma4(a, b, acc0); }
    { const float* p = pk + (nt0 + 1) * HW; v2f b;
      b.x = __builtin_nontemporal_load(p); b.y = __builtin_nontemporal_load(p + CHS);
      acc1 = wmma4(a, b, acc1); }
    { const float* p = pk + (nt0 + 2) * HW; v2f b;
      b.x = __builtin_nontemporal_load(p); b.y = __builtin_nontemporal_load(p + CHS);
      acc2 = wmma4(a, b, acc2); }
    { const float* p = pk + (nt0 + 3) * HW; v2f b;
      b.x = __builtin_nontemporal_load(p); b.y = __builtin_nontemporal_load(p + CHS);
      acc3 = wmma4(a, b, acc3); }
  }
  {
    float bv[8];
    #pragma unroll
    for (int r = 0; r < 8; ++r) bv[r] = bt[r + 8 * hl];
    #pragma unroll
    for (int r = 0; r < 8; ++r) {
      const int m = r + 8 * hl;                 // C/D layout: M = r + 8*(lane>=16)
      dst[m * 256 + (nt0 + 0) * 16 + ln] = acc0[r] + bv[r];
      dst[m * 256 + (nt0 + 1) * 16 + ln] = acc1[r] + bv[r];
      dst[m * 256 + (nt0 + 2) * 16 + ln] = acc2[r] + bv[r];
      dst[m * 256 + (nt0 + 3) * 16 + ln] = acc3[r] + bv[r];
    }
  }
  __syncthreads();

  // ================= Phase B: adaptive avg-pool 16->7 (overlapping windows) =
  #pragma unroll 1
  for (int idx = tid; idx < 2 * K49 * CMID; idx += 256) {
    const int half = (idx >= K49 * CMID) ? 1 : 0;
    const int rem  = half ? idx - K49 * CMID : idx;
    const int cch  = rem / K49;
    const int ij   = rem % K49;
    const int i = ij / 7, j = ij % 7;
    const int si = (i * 16) / 7, ei = ((i + 1) * 16 + 6) / 7;
    const int sj = (j * 16) / 7, ej = ((j + 1) * 16 + 6) / 7;
    const float* src = (half ? v_lds : f_lds) + cch * 256;
    float s = 0.f;
    for (int wi = si; wi < ei; ++wi)
      for (int hj = sj; hj < ej; ++hj)
        s += src[wi * 16 + hj];
    s /= (float)((ei - si) * (ej - sj));
    if (half) aggb[ij * CMID + cch] = s;    // value_centers seeds agg numerator
    else      cent[ij * CMID + cch] = s;
  }
  if (tid < K49) denom[tid] = 1.0f;         // (+1.0 in reference denominator)
  __syncthreads();

  // ================= Phase C1: normalize centers (rows of 16) ==============
  if (tid < K49) {
    float nrm = 0.f;
    #pragma unroll
    for (int c = 0; c < CMID; ++c) { const float u = cent[tid * CMID + c]; nrm += u * u; }
    const float inv = 1.0f / fmaxf(sqrtf(nrm), 1e-12f);
    #pragma unroll
    for (int c = 0; c < CMID; ++c) cent[tid * CMID + c] *= inv;
  }
  __syncthreads();

  // ========= Phase C2: per-position cosine sim, argmax, scatter-add ========
  const float alpha = alpha_p[0];
  const float beta  = beta_p[0];
  float fc[CMID];
  float nrm = 0.f;
  #pragma unroll
  for (int c = 0; c < CMID; ++c) { fc[c] = f_lds[c * 256 + tid]; nrm += fc[c] * fc[c]; }
  const float finv = 1.0f / fmaxf(sqrtf(nrm), 1e-12f);
  float zbest = -__builtin_inff();
  int   bk    = 0;
  #pragma unroll 1
  for (int k = 0; k < K49; ++k) {
    float d = 0.f;
    #pragma unroll
    for (int c = 0; c < CMID; ++c) d += cent[k * CMID + c] * fc[c];   // LDS broadcast
    const float z = alpha * (d * finv);     // sigmoid monotone -> argmax on z
    if (z > zbest) { zbest = z; bk = k; }
  }
  const float sn = 1.0f / (1.0f + __expf(-(beta + zbest)));
  #pragma unroll
  for (int c = 0; c < CMID; ++c)
    atomicAdd(&aggb[bk * CMID + c], sn * v_lds[c * 256 + tid]);   // ds_add_f32
  atomicAdd(&denom[bk], sn);
  __syncthreads();

  // ================= Phase C3: agg /= denom ================================
  #pragma unroll 1
  for (int idx = tid; idx < K49 * CMID; idx += 256)
    aggb[idx] = aggb[idx] / denom[idx >> 4];
  __syncthreads();

  // ======== Phase C4: out[n] = s_n * agg[k_n]  -> reuse f_lds (16x256) =====
  #pragma unroll
  for (int c = 0; c < CMID; ++c)
    f_lds[c * 256 + tid] = sn * aggb[bk * CMID + c];
  __syncthreads();

  // ====== Phase E: final = p_w(256x16) @ out(16x256) + p_b  (WMMA f32) =====
  float* og = outg + (long)batch * 256 * CHS + base;
  const float* o_lds = f_lds;
  #pragma unroll 1
  for (int mm = 0; mm < 2; ++mm) {
    const int m_tile = wave * 2 + mm;
    // A frags: p_w row-major (256 x 16); lane row o = m_tile*16+ln
    const float* pwp = p_w + (m_tile * 16 + ln) * CMID + 2 * hl;
    const v2f a0 = *(const v2f*)(pwp + 0);
    const v2f a1 = *(const v2f*)(pwp + 4);
    const v2f a2 = *(const v2f*)(pwp + 8);
    const v2f a3 = *(const v2f*)(pwp + 12);
    float pb[8];
    #pragma unroll
    for (int r = 0; r < 8; ++r) pb[r] = p_b[m_tile * 16 + r + 8 * hl];
    #pragma unroll 1
    for (int n_tile = 0; n_tile < 16; ++n_tile) {
      v8f acc = {};
      const float* bp = o_lds + n_tile * 16 + ln + 2 * hl * 256;
      { v2f b; b.x = bp[0];    b.y = bp[256];  acc = wmma4(a0, b, acc); }
      { v2f b; b.x = bp[1024]; b.y = bp[1280]; acc = wmma4(a1, b, acc); }
      { v2f b; b.x = bp[2048]; b.y = bp[2304]; acc = wmma4(a2, b, acc); }
      { v2f b; b.x = bp[3072]; b.y = bp[3328]; acc = wmma4(a3, b, acc); }
      float* orow = og + n_tile * HW + ln;
      #pragma unroll
      for (int r = 0; r < 8; ++r) {
        const int o = m_tile * 16 + r + 8 * hl;
        // write-once streaming output -> non-temporal, 64B-coalesced per half-wave
        __builtin_nontemporal_store(acc[r] + pb[r], orow + (long)o * CHS);
      }
    }
  }
}

extern "C" void kernel_launch(void* const* d_in, const int* in_sizes, int n_in,
                              void* d_out, int out_size, void* d_ws, size_t ws_size,
                              hipStream_t stream) {
  const float* x   = (const float*)d_in[0];
  const float* f_w = (const float*)d_in[1];
  const float* f_b = (const float*)d_in[2];
  const float* v_w = (const float*)d_in[3];
  const float* v_b = (const float*)d_in[4];
  const float* p_w = (const float*)d_in[5];
  const float* p_b = (const float*)d_in[6];
  const float* al  = (const float*)d_in[7];
  const float* be  = (const float*)d_in[8];
  float* out = (float*)d_out;
  (void)in_sizes; (void)n_in; (void)out_size; (void)d_ws; (void)ws_size;

  dim3 grid(16 * 49);   // 784 tiles, one block each
  dim3 block(256);      // 8 waves (wave32)
  local_cluster_fused<<<grid, block, 0, stream>>>(x, f_w, f_b, v_w, v_b,
                                                  p_w, p_b, al, be, out);
}